// SimpleAttentionLayer_84164179132466
// MI455X (gfx1250) — compile-verified
//
#include <hip/hip_runtime.h>
#include <math.h>

// CDNA5 wave32: v2f = per-lane slice of a 16x4 (or 4x16) f32 WMMA operand,
// v8f = per-lane slice of the 16x16 f32 accumulator.
typedef __attribute__((ext_vector_type(2))) float v2f;
typedef __attribute__((ext_vector_type(8))) float v8f;

#define B_       8
#define H_       4096
#define NH_      32
#define HD_      128
#define MAXSEQ_  2048
#define THREEH_  (3 * H_)

// ---------------------------------------------------------------------------
// out[8][N] = A8[8][K] @ W[K][N] + bias[N]
// One block per 16-column tile. 8 waves split K (split-K in registers),
// each wave chains V_WMMA_F32_16X16X4_F32; LDS cross-wave reduction.
// A matrix rows 8..15 are zero-padded via a 0/1 lane mask (no exec
// divergence in the hot loop -> unconditional b64 loads).
// ---------------------------------------------------------------------------
__global__ __launch_bounds__(256) void gemm8_wmma_kernel(
    const float* __restrict__ A8, const float* __restrict__ W,
    const float* __restrict__ bias, float* __restrict__ out,
    int K, int N)
{
    __shared__ float red[8 * 256];   // [wave][r][lane]

    const int tid  = threadIdx.x;
    const int wave = tid >> 5;
    const int lane = tid & 31;
    const int col0 = blockIdx.x * 16;

    const int kchunk = K >> 3;        // K split across 8 waves
    const int kbase  = wave * kchunk;

    // ISA A-layout (16x4 f32): lanes 0-15 -> K = k+0,k+1 ; lanes 16-31 -> K = k+2,k+3
    const int   mrow  = lane & 15;
    const int   kA    = (lane < 16) ? 0 : 2;
    // ISA B-layout (4x16 f32): lanes 0-15 -> rows k+0,k+1 ; lanes 16-31 -> rows k+2,k+3
    const int   nB    = lane & 15;
    const int   kB    = (lane < 16) ? 0 : 2;
    const float amask = (mrow < B_) ? 1.0f : 0.0f;   // zero-pad rows 8..15

    // Unconditional, pointer-stepped addressing in the hot loop.
    const float* ap = A8 + (size_t)(mrow & 7) * K + kbase + kA;          // float2 per step
    const float* bp = W + (size_t)(kbase + kB) * N + col0 + nB;          // bp[0], bp[N]
    const size_t bstep = 4 * (size_t)N;

    v8f c = {};
    const int iters = kchunk >> 2;    // 4 K per WMMA
#pragma unroll 4
    for (int it = 0; it < iters; ++it) {
        float2 av = *(const float2*)ap;        // contiguous pair -> global_load_b64
        v2f a, b;
        a.x = av.x * amask;
        a.y = av.y * amask;
        b.x = bp[0];
        b.y = bp[N];
        c = __builtin_amdgcn_wmma_f32_16x16x4_f32(
                /*neg_a=*/false, a, /*neg_b=*/false, b,
                /*c_mod=*/(short)0, c, /*reuse_a=*/false, /*reuse_b=*/false);
        ap += 4;
        bp += bstep;
    }

#pragma unroll
    for (int r = 0; r < 8; ++r) red[wave * 256 + r * 32 + lane] = c[r];
    __syncthreads();

    float sum = 0.0f;
#pragma unroll
    for (int w = 0; w < 8; ++w) sum += red[w * 256 + tid];

    // Decode C layout: element (r = tid>>5, lane = tid&31) -> M = r + (lane<16 ? 0:8)
    const int r  = tid >> 5;
    const int ln = tid & 31;
    const int M  = r + ((ln < 16) ? 0 : 8);
    const int Nc = col0 + (ln & 15);
    if (M < B_) out[M * N + Nc] = sum + bias[Nc];
}

// ---------------------------------------------------------------------------
// Single-token attention for one (b, h) per block. 256 threads = 8 waves.
// Row `cache_pos` of the K/V caches is logically replaced by the fresh k/v
// coming out of the QKV GEMM (reference semantics).
// ---------------------------------------------------------------------------
__global__ __launch_bounds__(256) void decode_attn_kernel(
    const float* __restrict__ qkv, const float* __restrict__ cache_k,
    const float* __restrict__ cache_v, const int* __restrict__ cache_pos_p,
    float* __restrict__ attn_out)
{
    __shared__ float4 lds_q[32];
    __shared__ float  lds_s[MAXSEQ_];
    __shared__ float  lds_red[8];
    __shared__ float4 lds_acc[8 * 32];

    const int tid  = threadIdx.x;
    const int wave = tid >> 5;
    const int lane = tid & 31;
    const int bh   = blockIdx.x;
    const int b    = bh / NH_;
    const int h    = bh % NH_;
    const int cp   = *cache_pos_p;
    const int L    = cp + 1;
    const float scale = 0.08838834764831845f;  // 1/sqrt(128)

    const float* qrow  = qkv + (size_t)b * THREEH_ +          h * HD_;
    const float* knew  = qkv + (size_t)b * THREEH_ + H_     + h * HD_;
    const float* vnew  = qkv + (size_t)b * THREEH_ + 2 * H_ + h * HD_;
    const float* kbase = cache_k + ((size_t)(b * NH_ + h)) * MAXSEQ_ * HD_;
    const float* vbase = cache_v + ((size_t)(b * NH_ + h)) * MAXSEQ_ * HD_;

    if (tid < 32) lds_q[tid] = ((const float4*)qrow)[tid];
    __syncthreads();

    // ---- scores: wave-cooperative dot(q, K[k]) -> lds_s[k] -----------------
    for (int k = wave; k < L; k += 8) {
        const float* krp = (k == cp) ? knew : (kbase + (size_t)k * HD_);
        if (k + 8 < L)
            __builtin_prefetch(kbase + (size_t)(k + 8) * HD_ + lane * 4, 0, 0);
        float4 kv = ((const float4*)krp)[lane];
        float4 qv = lds_q[lane];
        float p = kv.x * qv.x + kv.y * qv.y + kv.z * qv.z + kv.w * qv.w;
#pragma unroll
        for (int off = 16; off > 0; off >>= 1) p += __shfl_xor(p, off, 32);
        if (lane == 0) lds_s[k] = p * scale;
    }
    __syncthreads();

    // ---- softmax over L entries in LDS ------------------------------------
    float lmax = -3.402823466e38f;
    for (int i = tid; i < L; i += 256) lmax = fmaxf(lmax, lds_s[i]);
#pragma unroll
    for (int off = 16; off > 0; off >>= 1) lmax = fmaxf(lmax, __shfl_xor(lmax, off, 32));
    if (lane == 0) lds_red[wave] = lmax;
    __syncthreads();
    float gmax = lds_red[0];
#pragma unroll
    for (int w = 1; w < 8; ++w) gmax = fmaxf(gmax, lds_red[w]);
    __syncthreads();

    float lsum = 0.0f;
    for (int i = tid; i < L; i += 256) {
        float e = __expf(lds_s[i] - gmax);
        lds_s[i] = e;
        lsum += e;
    }
#pragma unroll
    for (int off = 16; off > 0; off >>= 1) lsum += __shfl_xor(lsum, off, 32);
    if (lane == 0) lds_red[wave] = lsum;
    __syncthreads();
    float gsum = 0.0f;
#pragma unroll
    for (int w = 0; w < 8; ++w) gsum += lds_red[w];
    const float inv = 1.0f / gsum;

    // ---- out[d] = sum_k p[k] * V[k][d] (coalesced float4 over d) ----------
    const int dg = lane;   // 32 x float4 covers HD=128
    const int kg = wave;   // 8 key groups
    float4 acc = make_float4(0.f, 0.f, 0.f, 0.f);
    for (int k = kg; k < L; k += 8) {
        const float* vrp = (k == cp) ? vnew : (vbase + (size_t)k * HD_);
        float4 vv = ((const float4*)vrp)[dg];
        float  p  = lds_s[k];
        acc.x += p * vv.x; acc.y += p * vv.y;
        acc.z += p * vv.z; acc.w += p * vv.w;
    }
    lds_acc[kg * 32 + dg] = acc;
    __syncthreads();
    if (kg == 0) {
        float4 t = lds_acc[dg];
#pragma unroll
        for (int w = 1; w < 8; ++w) {
            float4 u = lds_acc[w * 32 + dg];
            t.x += u.x; t.y += u.y; t.z += u.z; t.w += u.w;
        }
        float* o = attn_out + (size_t)b * H_ + h * HD_ + dg * 4;
        o[0] = t.x * inv; o[1] = t.y * inv; o[2] = t.z * inv; o[3] = t.w * inv;
    }
}

// ---------------------------------------------------------------------------
extern "C" void kernel_launch(void* const* d_in, const int* in_sizes, int n_in,
                              void* d_out, int out_size, void* d_ws, size_t ws_size,
                              hipStream_t stream) {
    (void)in_sizes; (void)n_in; (void)out_size; (void)ws_size;
    const float* x      = (const float*)d_in[0];
    const float* ck     = (const float*)d_in[1];
    const float* cv     = (const float*)d_in[2];
    const float* W_qkv  = (const float*)d_in[3];
    const float* b_qkv  = (const float*)d_in[4];
    const float* W_proj = (const float*)d_in[5];
    const float* b_proj = (const float*)d_in[6];
    const int*   cpos   = (const int*)d_in[7];
    float* out  = (float*)d_out;

    float* qkv  = (float*)d_ws;               // 8 * 12288 floats
    float* attn = qkv + (size_t)B_ * THREEH_; // 8 * 4096 floats

    // 1) QKV projection: 768 column tiles of 16
    gemm8_wmma_kernel<<<THREEH_ / 16, 256, 0, stream>>>(x, W_qkv, b_qkv, qkv, H_, THREEH_);
    // 2) attention: one block per (b, h)
    decode_attn_kernel<<<B_ * NH_, 256, 0, stream>>>(qkv, ck, cv, cpos, attn);
    // 3) output projection: 256 column tiles of 16
    gemm8_wmma_kernel<<<H_ / 16, 256, 0, stream>>>(attn, W_proj, b_proj, out, H_, H_);
}